// ArrdecoderModel_45380624450329
// MI455X (gfx1250) — compile-verified
//
#include <hip/hip_runtime.h>

#define B_  64
#define W_  4096
#define H_  128
#define I_  128
#define NC_ 32        // context W-chunks
#define AP  132       // LDS row pitch (132 % 64 == 4 -> conflict-free column reads)

typedef __attribute__((ext_vector_type(2))) float v2f;
typedef __attribute__((ext_vector_type(8))) float v8f;

__device__ __forceinline__ float sigf(float x) { return 1.0f / (1.0f + __expf(-x)); }

// ---------------------------------------------------------------------------
// att bias: bias[b] = sum_h (h0[L-1][b][h]*c0[L-1][b][h]) * att_W[H+h] + att_b
// ---------------------------------------------------------------------------
__global__ __launch_bounds__(128) void k_attbias(const float* __restrict__ h0,
                                                 const float* __restrict__ c0,
                                                 const float* __restrict__ att_W,
                                                 const float* __restrict__ att_b,
                                                 float* __restrict__ bias) {
  __shared__ float red[128];
  const int b = blockIdx.x, t = threadIdx.x;
  float v = h0[B_ * H_ + b * H_ + t] * c0[B_ * H_ + b * H_ + t] * att_W[H_ + t];
  red[t] = v;
  __syncthreads();
  for (int s = 64; s > 0; s >>= 1) {
    if (t < s) red[t] += red[t + s];
    __syncthreads();
  }
  if (t == 0) bias[b] = red[0] + att_b[0];
}

// ---------------------------------------------------------------------------
// scores[b][w] = dot(enc[b][w][:], att_W[0:H]) + bias[b]
// one wave per row: lane loads float4 (512B coalesced), shuffle-reduce
// ---------------------------------------------------------------------------
__global__ __launch_bounds__(256) void k_scores(const float* __restrict__ enc,
                                                const float* __restrict__ att_W,
                                                const float* __restrict__ bias,
                                                float* __restrict__ scores) {
  const int b = blockIdx.y;
  const int wbase = blockIdx.x * 64;
  const int wv = threadIdx.x >> 5, lane = threadIdx.x & 31;
  const float4 wa = reinterpret_cast<const float4*>(att_W)[lane];
  const float bb = bias[b];
#pragma unroll
  for (int i = 0; i < 8; ++i) {
    const int w = wbase + wv * 8 + i;
    const float4 e =
        reinterpret_cast<const float4*>(enc + ((size_t)b * W_ + w) * H_)[lane];
    float p = e.x * wa.x + e.y * wa.y + e.z * wa.z + e.w * wa.w;
    p += __shfl_down(p, 16, 32);
    p += __shfl_down(p, 8, 32);
    p += __shfl_down(p, 4, 32);
    p += __shfl_down(p, 2, 32);
    p += __shfl_down(p, 1, 32);
    if (lane == 0) scores[b * W_ + w] = p + bb;
  }
}

// ---------------------------------------------------------------------------
// softmax stats over logits [0, scores[b][0..W-2]]
// ---------------------------------------------------------------------------
__global__ __launch_bounds__(256) void k_smstats(const float* __restrict__ scores,
                                                 float* __restrict__ mstat,
                                                 float* __restrict__ sstat) {
  __shared__ float red[256];
  const int b = blockIdx.x, t = threadIdx.x;
  float m = 0.0f;  // the implicit zero logit
  for (int w = t; w < W_ - 1; w += 256) m = fmaxf(m, scores[b * W_ + w]);
  red[t] = m;
  __syncthreads();
  for (int s = 128; s > 0; s >>= 1) {
    if (t < s) red[t] = fmaxf(red[t], red[t + s]);
    __syncthreads();
  }
  m = red[0];
  __syncthreads();
  float sum = (t == 0) ? __expf(-m) : 0.0f;
  for (int w = t; w < W_ - 1; w += 256) sum += __expf(scores[b * W_ + w] - m);
  red[t] = sum;
  __syncthreads();
  for (int s = 128; s > 0; s >>= 1) {
    if (t < s) red[t] += red[t + s];
    __syncthreads();
  }
  if (t == 0) {
    mstat[b] = m;
    sstat[b] = red[0];
  }
}

// ---------------------------------------------------------------------------
// context partials: partial[b][c][h] = sum_{w in chunk c} weight[b][w]*enc[b][w][h]
// ---------------------------------------------------------------------------
__global__ __launch_bounds__(128) void k_context(const float* __restrict__ enc,
                                                 const float* __restrict__ scores,
                                                 const float* __restrict__ mstat,
                                                 const float* __restrict__ sstat,
                                                 float* __restrict__ partial) {
  __shared__ float wts[W_ / NC_];
  const int b = blockIdx.y, c = blockIdx.x, t = threadIdx.x;
  const float m = mstat[b], inv = 1.0f / sstat[b];
  const int w = c * (W_ / NC_) + t;
  const float logit = (w == 0) ? 0.0f : scores[b * W_ + w - 1];
  wts[t] = __expf(logit - m) * inv;
  __syncthreads();
  float acc = 0.0f;
  const float* ebase = enc + ((size_t)b * W_ + c * (W_ / NC_)) * H_;
#pragma unroll 4
  for (int wl = 0; wl < W_ / NC_; ++wl) acc += wts[wl] * ebase[wl * H_ + t];
  partial[(b * NC_ + c) * H_ + t] = acc;
}

__global__ __launch_bounds__(256) void k_xreduce(const float* __restrict__ partial,
                                                 float* __restrict__ xbuf) {
  const int idx = blockIdx.x * 256 + threadIdx.x;  // B_*H_ = 8192 total
  const int b = idx >> 7, h = idx & 127;
  float s = 0.0f;
#pragma unroll
  for (int c = 0; c < NC_; ++c) s += partial[(b * NC_ + c) * H_ + h];
  xbuf[idx] = s;
}

// ---------------------------------------------------------------------------
// Tail: xin GEMM + two LSTM cells, all via V_WMMA_F32_16X16X4_F32.
// Single workgroup, 8 waves. Fragment layouts per ISA 7.12.2:
//   ln = lane&15, kh = lane>>4
//   A(16x4):  a = {A[m0+ln][k0+2kh], A[m0+ln][k0+2kh+1]}
//   B(4x16):  b = {B[k0+2kh][n0+ln], B[k0+2kh+1][n0+ln]}
//   C/D:      c[r] <-> D[m0 + r + 8*kh][n0 + ln]
// ---------------------------------------------------------------------------
__global__ __launch_bounds__(256) void k_tail(
    const float* __restrict__ xbuf, const float* __restrict__ input,
    const float* __restrict__ h0, const float* __restrict__ c0,
    const float* __restrict__ inp_W, const float* __restrict__ inp_b,
    const float* __restrict__ W_ih0, const float* __restrict__ W_hh0,
    const float* __restrict__ b_ih0, const float* __restrict__ b_hh0,
    const float* __restrict__ W_ih1, const float* __restrict__ W_hh1,
    const float* __restrict__ b_ih1, const float* __restrict__ b_hh1,
    float* __restrict__ out) {
  __shared__ float sA[B_ * AP];  // holds xin (phase A->B) then h1 (phase B->C)
  const int wv = threadIdx.x >> 5;
  const int lane = threadIdx.x & 31;
  const int ln = lane & 15;
  const int kh = lane >> 4;

  // ---- Phase A: xin = [x, input] @ inp_W^T + inp_b   (M=64,N=128,K=256)
  for (int tl = wv * 4; tl < wv * 4 + 4; ++tl) {
    const int m0 = (tl >> 3) * 16, n0 = (tl & 7) * 16;
    const float bb = inp_b[n0 + ln];
    v8f c;
#pragma unroll
    for (int r = 0; r < 8; ++r) c[r] = bb;
    const int arow = m0 + ln, bcol = n0 + ln;
    for (int k0 = 0; k0 < 2 * I_; k0 += 4) {
      const int ka = k0 + 2 * kh;
      float a0, a1;
      if (ka < I_) { a0 = xbuf[arow * I_ + ka]; a1 = xbuf[arow * I_ + ka + 1]; }
      else         { a0 = input[arow * I_ + ka - I_]; a1 = input[arow * I_ + ka - I_ + 1]; }
      v2f a = {a0, a1};
      v2f b = {inp_W[bcol * (2 * I_) + ka], inp_W[bcol * (2 * I_) + ka + 1]};
      c = __builtin_amdgcn_wmma_f32_16x16x4_f32(false, a, false, b, (short)0, c,
                                                false, false);
    }
#pragma unroll
    for (int r = 0; r < 8; ++r) sA[(m0 + r + 8 * kh) * AP + n0 + ln] = c[r];
  }
  __syncthreads();

  // ---- Phase B: layer-0 LSTM gates + activations
  v8f hsave[4];
  int msave[4], nsave[4];
  for (int p = 0; p < 4; ++p) {
    const int tl = wv * 4 + p;
    const int m0 = (tl >> 3) * 16, n0 = (tl & 7) * 16;
    v8f acc[4];
#pragma unroll
    for (int g = 0; g < 4; ++g) {
      const float bb = b_ih0[g * H_ + n0 + ln] + b_hh0[g * H_ + n0 + ln];
#pragma unroll
      for (int r = 0; r < 8; ++r) acc[g][r] = bb;
    }
    const int arow = m0 + ln;
    for (int k0 = 0; k0 < I_; k0 += 4) {
      const int ka = k0 + 2 * kh;
      v2f a = {sA[arow * AP + ka], sA[arow * AP + ka + 1]};
#pragma unroll
      for (int g = 0; g < 4; ++g) {
        const int wrow = g * H_ + n0 + ln;
        v2f b = {W_ih0[wrow * I_ + ka], W_ih0[wrow * I_ + ka + 1]};
        acc[g] = __builtin_amdgcn_wmma_f32_16x16x4_f32(false, a, false, b,
                                                       (short)0, acc[g], false, false);
      }
    }
    for (int k0 = 0; k0 < H_; k0 += 4) {
      const int ka = k0 + 2 * kh;
      v2f a = {h0[arow * H_ + ka], h0[arow * H_ + ka + 1]};  // layer 0
#pragma unroll
      for (int g = 0; g < 4; ++g) {
        const int wrow = g * H_ + n0 + ln;
        v2f b = {W_hh0[wrow * H_ + ka], W_hh0[wrow * H_ + ka + 1]};
        acc[g] = __builtin_amdgcn_wmma_f32_16x16x4_f32(false, a, false, b,
                                                       (short)0, acc[g], false, false);
      }
    }
    v8f hn;
#pragma unroll
    for (int r = 0; r < 8; ++r) {
      const int bi = m0 + r + 8 * kh, hi = n0 + ln;
      const float c0v = c0[bi * H_ + hi];  // layer 0
      const float iv = sigf(acc[0][r]);
      const float fv = sigf(acc[1][r]);
      const float gv = tanhf(acc[2][r]);
      const float ov = sigf(acc[3][r]);
      const float cn = fv * c0v + iv * gv;
      const float hv = ov * tanhf(cn);
      hn[r] = hv;
      out[8192 + bi * H_ + hi] = hv;   // h-stack layer 0
      out[24576 + bi * H_ + hi] = cn;  // c-stack layer 0
    }
    hsave[p] = hn; msave[p] = m0; nsave[p] = n0;
  }
  __syncthreads();  // all reads of xin complete
  for (int p = 0; p < 4; ++p) {
#pragma unroll
    for (int r = 0; r < 8; ++r)
      sA[(msave[p] + r + 8 * kh) * AP + nsave[p] + ln] = hsave[p][r];
  }
  __syncthreads();

  // ---- Phase C: layer-1 LSTM
  for (int p = 0; p < 4; ++p) {
    const int tl = wv * 4 + p;
    const int m0 = (tl >> 3) * 16, n0 = (tl & 7) * 16;
    v8f acc[4];
#pragma unroll
    for (int g = 0; g < 4; ++g) {
      const float bb = b_ih1[g * H_ + n0 + ln] + b_hh1[g * H_ + n0 + ln];
#pragma unroll
      for (int r = 0; r < 8; ++r) acc[g][r] = bb;
    }
    const int arow = m0 + ln;
    for (int k0 = 0; k0 < H_; k0 += 4) {
      const int ka = k0 + 2 * kh;
      v2f a = {sA[arow * AP + ka], sA[arow * AP + ka + 1]};  // h1
#pragma unroll
      for (int g = 0; g < 4; ++g) {
        const int wrow = g * H_ + n0 + ln;
        v2f b = {W_ih1[wrow * H_ + ka], W_ih1[wrow * H_ + ka + 1]};
        acc[g] = __builtin_amdgcn_wmma_f32_16x16x4_f32(false, a, false, b,
                                                       (short)0, acc[g], false, false);
      }
    }
    for (int k0 = 0; k0 < H_; k0 += 4) {
      const int ka = k0 + 2 * kh;
      v2f a = {h0[B_ * H_ + arow * H_ + ka], h0[B_ * H_ + arow * H_ + ka + 1]};  // layer 1
#pragma unroll
      for (int g = 0; g < 4; ++g) {
        const int wrow = g * H_ + n0 + ln;
        v2f b = {W_hh1[wrow * H_ + ka], W_hh1[wrow * H_ + ka + 1]};
        acc[g] = __builtin_amdgcn_wmma_f32_16x16x4_f32(false, a, false, b,
                                                       (short)0, acc[g], false, false);
      }
    }
#pragma unroll
    for (int r = 0; r < 8; ++r) {
      const int bi = m0 + r + 8 * kh, hi = n0 + ln;
      const float c0v = c0[B_ * H_ + bi * H_ + hi];  // layer 1
      const float iv = sigf(acc[0][r]);
      const float fv = sigf(acc[1][r]);
      const float gv = tanhf(acc[2][r]);
      const float ov = sigf(acc[3][r]);
      const float cn = fv * c0v + iv * gv;
      const float hv = ov * tanhf(cn);
      out[bi * H_ + hi] = hv;          // output (h2)
      out[16384 + bi * H_ + hi] = hv;  // h-stack layer 1
      out[32768 + bi * H_ + hi] = cn;  // c-stack layer 1
    }
  }
}

// ---------------------------------------------------------------------------
extern "C" void kernel_launch(void* const* d_in, const int* in_sizes, int n_in,
                              void* d_out, int out_size, void* d_ws, size_t ws_size,
                              hipStream_t stream) {
  const float* input  = (const float*)d_in[0];
  const float* h0     = (const float*)d_in[1];
  const float* c0     = (const float*)d_in[2];
  const float* enc    = (const float*)d_in[3];
  const float* att_W  = (const float*)d_in[4];
  const float* att_b  = (const float*)d_in[5];
  const float* inp_W  = (const float*)d_in[6];
  const float* inp_b  = (const float*)d_in[7];
  const float* W_ih0  = (const float*)d_in[8];
  const float* W_hh0  = (const float*)d_in[9];
  const float* b_ih0  = (const float*)d_in[10];
  const float* b_hh0  = (const float*)d_in[11];
  const float* W_ih1  = (const float*)d_in[12];
  const float* W_hh1  = (const float*)d_in[13];
  const float* b_ih1  = (const float*)d_in[14];
  const float* b_hh1  = (const float*)d_in[15];
  float* out = (float*)d_out;

  float* ws      = (float*)d_ws;
  float* bias    = ws;            // 64
  float* mstat   = ws + 64;       // 64
  float* sstat   = ws + 128;      // 64
  float* xbuf    = ws + 256;      // 8192
  float* scores  = ws + 256 + 8192;                 // B*W = 262144
  float* partial = scores + (size_t)B_ * W_;        // B*NC*H = 262144

  k_attbias<<<B_, 128, 0, stream>>>(h0, c0, att_W, att_b, bias);
  k_scores<<<dim3(W_ / 64, B_), 256, 0, stream>>>(enc, att_W, bias, scores);
  k_smstats<<<B_, 256, 0, stream>>>(scores, mstat, sstat);
  k_context<<<dim3(NC_, B_), 128, 0, stream>>>(enc, scores, mstat, sstat, partial);
  k_xreduce<<<B_ * H_ / 256, 256, 0, stream>>>(partial, xbuf);
  k_tail<<<1, 256, 0, stream>>>(xbuf, input, h0, c0, inp_W, inp_b, W_ih0, W_hh0,
                                b_ih0, b_hh0, W_ih1, W_hh1, b_ih1, b_hh1, out);
}